// Set2Set_55405078118492
// MI455X (gfx1250) — compile-verified
//
#include <hip/hip_runtime.h>
#include <math.h>

#define NN 2000000
#define DD 128
#define BB 1024
#define STEPS 3
#define NEG_BIG (-3.402823466e38f)

typedef __attribute__((ext_vector_type(2))) float v2f;
typedef __attribute__((ext_vector_type(8))) float v8f;

__device__ __forceinline__ float sigmoidf_(float x) {
    return 1.0f / (1.0f + __expf(-x));
}

// ---------------- zero init (h, c, q_star are contiguous in ws) ----------------
__global__ void zero_f32(float* __restrict__ p, int n) {
    int i = blockIdx.x * blockDim.x + threadIdx.x;
    if (i < n) p[i] = 0.0f;
}

// ---------------- segment boundaries from sorted batch_index ----------------
__global__ void seg_bounds(const int* __restrict__ bi, int* __restrict__ starts) {
    int n = blockIdx.x * blockDim.x + threadIdx.x;
    if (n >= NN) return;
    int b = bi[n];
    if (n == 0) {
        for (int j = 0; j <= b; ++j) starts[j] = 0;
    } else {
        int pb = bi[n - 1];
        for (int j = pb + 1; j <= b; ++j) starts[j] = n;
    }
    if (n == NN - 1) {
        for (int j = b + 1; j <= BB; ++j) starts[j] = NN;
    }
}

// ---------------- LSTM cell via V_WMMA_F32_16X16X4_F32 ----------------
// gates[b,n] = sum_k qs[b,k]*Wih[n,k] + sum_k h[b,k]*Whh[n,k] + bih[n] + bhh[n]
// Block: 256 threads (8 waves). Block covers 16 batch rows x all 512 gate cols.
// Each wave owns 4 N-tiles of 16 columns. f32 WMMA: A 16x4 (v2f/lane), B 4x16
// (v2f/lane), C/D 16x16 (v8f/lane).
__global__ void __launch_bounds__(256)
lstm_wmma(const float* __restrict__ qs,   // [B, 2D]
          const float* __restrict__ Wih,  // [4D, 2D] row-major
          const float* __restrict__ Whh,  // [4D, D]  row-major
          const float* __restrict__ bih,  // [4D]
          const float* __restrict__ bhh,  // [4D]
          float* __restrict__ h,          // [B, D] (in: h_t, out: h_{t+1})
          float* __restrict__ c)          // [B, D] (in/out)
{
    __shared__ float G[16][4 * DD];       // 16 x 512 gates, 32 KB

    const int b0    = blockIdx.x * 16;
    const int wave  = threadIdx.x >> 5;
    const int lane  = threadIdx.x & 31;
    const int l15   = lane & 15;          // A: row m ; B: col n (within tile)
    const int khalf = (lane >> 4) * 2;    // K pair selected by lane half
    const int nbase = wave * 64;

    const v8f vzero = {0.f, 0.f, 0.f, 0.f, 0.f, 0.f, 0.f, 0.f};
    v8f acc[4] = {vzero, vzero, vzero, vzero};

    // ---- K over q_star (2D = 256) with W_ih ----
    const float* arow0 = qs + (size_t)(b0 + l15) * (2 * DD);
    for (int k0 = 0; k0 < 2 * DD; k0 += 4) {
        v2f a = *(const v2f*)(arow0 + k0 + khalf);
#pragma unroll
        for (int t = 0; t < 4; ++t) {
            const float* brow = Wih + (size_t)(nbase + t * 16 + l15) * (2 * DD);
            v2f bf = *(const v2f*)(brow + k0 + khalf);
            acc[t] = __builtin_amdgcn_wmma_f32_16x16x4_f32(
                false, a, false, bf, (short)0, acc[t], false, false);
        }
    }

    // ---- K over h (D = 128) with W_hh ----
    const float* arow1 = h + (size_t)(b0 + l15) * DD;
    for (int k0 = 0; k0 < DD; k0 += 4) {
        v2f a = *(const v2f*)(arow1 + k0 + khalf);
#pragma unroll
        for (int t = 0; t < 4; ++t) {
            const float* brow = Whh + (size_t)(nbase + t * 16 + l15) * DD;
            v2f bf = *(const v2f*)(brow + k0 + khalf);
            acc[t] = __builtin_amdgcn_wmma_f32_16x16x4_f32(
                false, a, false, bf, (short)0, acc[t], false, false);
        }
    }

    // ---- spill gates (+ biases) to LDS ----
    const int mbase = (lane >> 4) * 8;    // C/D: lanes 16-31 hold rows 8..15
#pragma unroll
    for (int t = 0; t < 4; ++t) {
        int n = nbase + t * 16 + l15;
        float bias = bih[n] + bhh[n];
#pragma unroll
        for (int r = 0; r < 8; ++r)
            G[mbase + r][n] = acc[t][r] + bias;
    }
    __syncthreads();   // all h reads above complete before h is overwritten below

    // ---- elementwise cell update: i,f,g,o gate order ----
    for (int idx = threadIdx.x; idx < 16 * DD; idx += 256) {
        int row = idx >> 7;
        int d   = idx & (DD - 1);
        int gi  = b0 + row;
        float ig = sigmoidf_(G[row][d]);
        float fg = sigmoidf_(G[row][d + DD]);
        float gg = tanhf(G[row][d + 2 * DD]);
        float og = sigmoidf_(G[row][d + 3 * DD]);
        float cn = fg * c[(size_t)gi * DD + d] + ig * gg;
        c[(size_t)gi * DD + d] = cn;
        h[(size_t)gi * DD + d] = og * tanhf(cn);
    }
}

// ---------------- fused attention + segment softmax + pooling ----------------
// One block per graph (contiguous node range). Single streaming pass over x
// with online softmax: per wave keep (m, sum, v[128]); combine 8 waves in LDS.
__global__ void __launch_bounds__(256)
attn_pool(const float* __restrict__ x,       // [N, D]
          const int*   __restrict__ starts,  // [B+1]
          const float* __restrict__ h,       // [B, D] (q)
          float*       __restrict__ qstar)   // [B, 2D] out
{
    const int b    = blockIdx.x;
    const int wave = threadIdx.x >> 5;
    const int lane = threadIdx.x & 31;
    const int s0 = starts[b];
    const int s1 = starts[b + 1];

    const float4 q4 = *(const float4*)(h + (size_t)b * DD + lane * 4);

    float m = NEG_BIG, ssum = 0.0f;
    float4 v = make_float4(0.f, 0.f, 0.f, 0.f);

    auto update = [&](const float4 xv) {
        float e = xv.x * q4.x + xv.y * q4.y + xv.z * q4.z + xv.w * q4.w;
        e += __shfl_xor(e, 16, 32);
        e += __shfl_xor(e, 8, 32);
        e += __shfl_xor(e, 4, 32);
        e += __shfl_xor(e, 2, 32);
        e += __shfl_xor(e, 1, 32);
        float mn = fmaxf(m, e);
        float scale = __expf(m - mn);   // 0 on first node (m = -BIG)
        float w = __expf(e - mn);
        ssum = ssum * scale + w;
        v.x = v.x * scale + w * xv.x;
        v.y = v.y * scale + w * xv.y;
        v.z = v.z * scale + w * xv.z;
        v.w = v.w * scale + w * xv.w;
        m = mn;
    };

    int n = s0 + wave;
    for (; n + 8 < s1; n += 16) {        // 2 nodes in flight per wave
        float4 x0 = *(const float4*)(x + (size_t)n * DD + lane * 4);
        float4 x1 = *(const float4*)(x + (size_t)(n + 8) * DD + lane * 4);
        update(x0);
        update(x1);
    }
    if (n < s1) {
        float4 x0 = *(const float4*)(x + (size_t)n * DD + lane * 4);
        update(x0);
    }

    __shared__ float sm[8], ss[8];
    __shared__ float sv[8][DD];
    if (lane == 0) { sm[wave] = m; ss[wave] = ssum; }
    sv[wave][lane * 4 + 0] = v.x;
    sv[wave][lane * 4 + 1] = v.y;
    sv[wave][lane * 4 + 2] = v.z;
    sv[wave][lane * 4 + 3] = v.w;
    __syncthreads();

    if (threadIdx.x < DD) {
        int j = threadIdx.x;
        float M = NEG_BIG;
#pragma unroll
        for (int w = 0; w < 8; ++w) M = fmaxf(M, sm[w]);
        float S = 0.f, V = 0.f;
#pragma unroll
        for (int w = 0; w < 8; ++w) {
            if (ss[w] > 0.f) {
                float f = __expf(sm[w] - M);
                S += ss[w] * f;
                V += sv[w][j] * f;
            }
        }
        float r = V / (S + 1e-16f);
        qstar[(size_t)b * (2 * DD) + j]      = h[(size_t)b * DD + j]; // q
        qstar[(size_t)b * (2 * DD) + DD + j] = r;                    // r
    }
}

// ---------------- launcher ----------------
extern "C" void kernel_launch(void* const* d_in, const int* in_sizes, int n_in,
                              void* d_out, int out_size, void* d_ws, size_t ws_size,
                              hipStream_t stream)
{
    const float* x   = (const float*)d_in[0];
    const int*   bi  = (const int*)d_in[1];
    const float* Wih = (const float*)d_in[2];
    const float* Whh = (const float*)d_in[3];
    const float* bih = (const float*)d_in[4];
    const float* bhh = (const float*)d_in[5];
    float* out = (float*)d_out;

    char* p = (char*)d_ws;
    int* starts = (int*)p;   p += ((4 * (BB + 1) + 255) / 256) * 256;
    float* h  = (float*)p;   p += (size_t)BB * DD * sizeof(float);
    float* c  = (float*)p;   p += (size_t)BB * DD * sizeof(float);
    float* qs = (float*)p;   p += (size_t)BB * 2 * DD * sizeof(float);

    // h, c, qs are contiguous: one zero pass (q_star must be 0 for step 1)
    const int ztot = BB * DD + BB * DD + BB * 2 * DD;
    zero_f32<<<(ztot + 255) / 256, 256, 0, stream>>>(h, ztot);

    seg_bounds<<<(NN + 255) / 256, 256, 0, stream>>>(bi, starts);

    for (int s = 0; s < STEPS; ++s) {
        lstm_wmma<<<BB / 16, 256, 0, stream>>>(qs, Wih, Whh, bih, bhh, h, c);
        attn_pool<<<BB, 256, 0, stream>>>(x, starts, h,
                                          (s == STEPS - 1) ? out : qs);
    }
}